// MultiScaleRoIAlign_83021717832162
// MI455X (gfx1250) — compile-verified
//
#include <hip/hip_runtime.h>
#include <math.h>

#define OUT_HW 7
#define C_CH   256
#define BINS   49   // 7*7
#define ROIS_PER_BATCH 512

typedef __attribute__((address_space(1))) int* as1_int_ptr;
typedef __attribute__((address_space(3))) int* as3_int_ptr;

__global__ __launch_bounds__(256) void msroi_align_kernel(
    const float* __restrict__ f0, const float* __restrict__ f1,
    const float* __restrict__ f2, const float* __restrict__ f3,
    const float* __restrict__ boxes, float* __restrict__ out)
{
    __shared__ float s_box[4];
    __shared__ int   s_lo[14];     // [0..6] x-axis, [7..13] y-axis
    __shared__ int   s_hi[14];
    __shared__ float s_fr[14];
    __shared__ int   s_valid[14];
    __shared__ const float* s_base; // feat + (b*C)*H*W for selected level
    __shared__ int   s_W;

    const int roi = blockIdx.x;
    const int tid = threadIdx.x;

    // ---- stage the 4 box floats into LDS (async-to-LDS path on CDNA5 if exposed) ----
#if defined(__gfx1250__) && __has_builtin(__builtin_amdgcn_global_load_async_to_lds_b32)
    if (tid < 4) {
        __builtin_amdgcn_global_load_async_to_lds_b32(
            (as1_int_ptr)(boxes + (size_t)roi * 4 + tid),
            (as3_int_ptr)&s_box[tid],
            0, 0);
    }
    if (tid < 32) {   // only wave 0 issued the async ops; SOPP wait is wave-level
  #if __has_builtin(__builtin_amdgcn_s_wait_asynccnt)
        __builtin_amdgcn_s_wait_asynccnt(0);
  #else
        asm volatile("s_wait_asynccnt 0" ::: "memory");
  #endif
    }
#else
    if (tid < 4) s_box[tid] = boxes[(size_t)roi * 4 + tid];
#endif
    __syncthreads();

    // ---- per-ROI setup: level selection + per-axis bilinear tables (threads 0..13) ----
    if (tid < 14) {
        const float bx1 = s_box[0], by1 = s_box[1], bx2 = s_box[2], by2 = s_box[3];
        // FPN level assignment (on raw coords)
        const float size = sqrtf((bx2 - bx1 + 1.0f) * (by2 - by1 + 1.0f));
        float lvf = floorf(4.0f + log2f(size * (1.0f / 224.0f)) + 1e-8f);
        lvf = fminf(fmaxf(lvf, 2.0f), 5.0f);
        const int   l     = (int)lvf - 2;            // 0..3
        const int   W     = 256 >> l;                // 256,128,64,32
        const float scale = 1.0f / (float)(4 << l);  // 1/4,1/8,1/16,1/32

        const int   axis = (tid >= 7) ? 1 : 0;       // 0 = x, 1 = y
        const int   p    = axis ? (tid - 7) : tid;
        const float c1   = (axis ? by1 : bx1) * scale;
        const float c2   = (axis ? by2 : bx2) * scale;
        const float roi_sz = fmaxf(c2 - c1, 1.0f);
        const float binsz  = roi_sz * (1.0f / OUT_HW);
        const float s      = c1 + ((float)p + 0.5f) * binsz;

        const float fsize = (float)W;
        const int valid   = (s >= -1.0f) && (s <= fsize);
        const float sc    = fmaxf(s, 0.0f);
        const int low_raw = (int)floorf(sc);
        const int hiflag  = (low_raw >= W - 1);
        s_lo[tid]    = hiflag ? (W - 1) : low_raw;
        s_hi[tid]    = hiflag ? (W - 1) : (low_raw + 1);
        s_fr[tid]    = hiflag ? 0.0f : (sc - (float)low_raw);
        s_valid[tid] = valid;

        if (tid == 0) {
            const float* f = (l == 0) ? f0 : (l == 1) ? f1 : (l == 2) ? f2 : f3;
            const int b = roi / ROIS_PER_BATCH;
            s_base = f + (size_t)(b * C_CH) * ((size_t)W * (size_t)W);
            s_W = W;
        }
    }
    __syncthreads();

    const float* __restrict__ base = s_base;
    const int W  = s_W;
    const int HW = W * W;
    float* __restrict__ outRoi = out + (size_t)roi * (size_t)(C_CH * BINS);

    // 256 ch * 49 bins = 12544 outputs per ROI -> 49 fully-coalesced iterations
    #pragma unroll 1
    for (int it = 0; it < (C_CH * BINS) / 256; ++it) {
        const int idx = it * 256 + tid;
        const int c   = idx / BINS;
        const int bin = idx - c * BINS;
        const int py  = bin / OUT_HW;
        const int px  = bin - py * OUT_HW;

        const int   xlo = s_lo[px],     xhi = s_hi[px];
        const int   ylo = s_lo[7 + py], yhi = s_hi[7 + py];
        const float fx  = s_fr[px],     fy  = s_fr[7 + py];
        const int   v   = s_valid[px] & s_valid[7 + py];

        const float* __restrict__ pch = base + (size_t)c * (size_t)HW;
        const int r0 = ylo * W, r1 = yhi * W;
        const float p00 = pch[r0 + xlo];
        const float p01 = pch[r0 + xhi];
        const float p10 = pch[r1 + xlo];
        const float p11 = pch[r1 + xhi];

        // pull lines for a channel ~6 ahead into cache (speculative, dropped if bad)
        if (bin == 0 && c + 6 < C_CH)
            __builtin_prefetch(pch + (size_t)6 * HW + r0 + xlo, 0, 1);

        const float hx = 1.0f - fx, hy = 1.0f - fy;
        float val = p00 * (hy * hx) + p01 * (hy * fx)
                  + p10 * (fy * hx) + p11 * (fy * fx);
        outRoi[idx] = v ? val : 0.0f;
    }
}

extern "C" void kernel_launch(void* const* d_in, const int* in_sizes, int n_in,
                              void* d_out, int out_size, void* d_ws, size_t ws_size,
                              hipStream_t stream) {
    (void)n_in; (void)out_size; (void)d_ws; (void)ws_size;
    const float* f0    = (const float*)d_in[0];
    const float* f1    = (const float*)d_in[1];
    const float* f2    = (const float*)d_in[2];
    const float* f3    = (const float*)d_in[3];
    const float* boxes = (const float*)d_in[4];
    float* out = (float*)d_out;

    const int nrois = in_sizes[4] / 4;   // N*L = 1024
    msroi_align_kernel<<<nrois, 256, 0, stream>>>(f0, f1, f2, f3, boxes, out);
}